// mLSTM_62654982914534
// MI455X (gfx1250) — compile-verified
//
#include <hip/hip_runtime.h>
#include <hip/hip_bf16.h>
#include <math.h>

#define D 4096
#define RSQRT_D 0.015625f  // 1/sqrt(4096)

typedef __attribute__((ext_vector_type(2))) float v2f;
typedef __attribute__((ext_vector_type(8))) float v8f;

// ---------------------------------------------------------------------------
// Kernel 1: four fused GEMVs (q, k, v, o) via V_WMMA_F32_16X16X4_F32.
// Grid: x = D/(16*4) row-tile groups, y = 4 (matrix select).
// Each wave owns a 16-row tile; loop over K in steps of 16 with 4 WMMA
// accumulators. A-matrix layout (16x4 f32): lanes 0-15 hold (row, K=0..1),
// lanes 16-31 hold (row-16's rows, K=2..3) -> one b64 load per lane.
// B is x[k0..k0+3] broadcast to all 16 N columns (same per-lane addressing).
// ---------------------------------------------------------------------------
__global__ __launch_bounds__(128) void k_gemv4_wmma(
    const float* __restrict__ x,
    const float* __restrict__ wq, const float* __restrict__ bq,
    const float* __restrict__ wk, const float* __restrict__ bk,
    const float* __restrict__ wv, const float* __restrict__ bv,
    const float* __restrict__ wo, const float* __restrict__ bo,
    float* __restrict__ out_q, float* __restrict__ out_k,
    float* __restrict__ out_v, float* __restrict__ out_o)
{
    const int lane = threadIdx.x & 31;
    const int wave = threadIdx.x >> 5;
    const int mat  = blockIdx.y;

    const float* W;
    const float* bias;
    float* out;
    switch (mat) {
        case 0:  W = wq; bias = bq; out = out_q; break;
        case 1:  W = wk; bias = bk; out = out_k; break;
        case 2:  W = wv; bias = bv; out = out_v; break;
        default: W = wo; bias = bo; out = out_o; break;
    }

    const int m0   = (blockIdx.x * 4 + wave) * 16;       // 16-row tile base
    const int row  = m0 + (lane & 15);
    const int koff = (lane >> 4) * 2;                    // 0 or 2

    const float* wrow = W + (size_t)row * D + koff;
    const float* xp   = x + koff;

    v8f c0 = {0.f,0.f,0.f,0.f,0.f,0.f,0.f,0.f};
    v8f c1 = {0.f,0.f,0.f,0.f,0.f,0.f,0.f,0.f};
    v8f c2 = {0.f,0.f,0.f,0.f,0.f,0.f,0.f,0.f};
    v8f c3 = {0.f,0.f,0.f,0.f,0.f,0.f,0.f,0.f};

    for (int k0 = 0; k0 < D; k0 += 16) {
        v2f a0 = *(const v2f*)(wrow + k0);
        v2f a1 = *(const v2f*)(wrow + k0 + 4);
        v2f a2 = *(const v2f*)(wrow + k0 + 8);
        v2f a3 = *(const v2f*)(wrow + k0 + 12);
        v2f b0 = *(const v2f*)(xp + k0);
        v2f b1 = *(const v2f*)(xp + k0 + 4);
        v2f b2 = *(const v2f*)(xp + k0 + 8);
        v2f b3 = *(const v2f*)(xp + k0 + 12);
        c0 = __builtin_amdgcn_wmma_f32_16x16x4_f32(false, a0, false, b0, (short)0, c0, false, false);
        c1 = __builtin_amdgcn_wmma_f32_16x16x4_f32(false, a1, false, b1, (short)0, c1, false, false);
        c2 = __builtin_amdgcn_wmma_f32_16x16x4_f32(false, a2, false, b2, (short)0, c2, false, false);
        c3 = __builtin_amdgcn_wmma_f32_16x16x4_f32(false, a3, false, b3, (short)0, c3, false, false);
    }
    v8f c = (c0 + c1) + (c2 + c3);

    // C/D layout: VGPR j -> M=j (lanes 0-15), M=8+j (lanes 16-31); every N
    // column holds the same GEMV value, so lanes 0 and 16 write the results.
    if ((lane & 15) == 0) {
        const int mofs = (lane >> 4) * 8;
        #pragma unroll
        for (int j = 0; j < 8; ++j) {
            const int m = m0 + mofs + j;
            float y = c[j] + bias[m];
            if (mat == 1) y *= RSQRT_D;                         // k scale
            if (mat == 3) y = 1.0f / (1.0f + __expf(-y));       // o sigmoid
            out[m] = y;
        }
    }
}

// ---------------------------------------------------------------------------
// Kernel 2 (single block): scalar gates i_t, f_t; n_t = f*n_prev + i*k;
// max_nqt = max(|n_t . q|, 1).  scal = {i_t, f_t, max_nqt}
// ---------------------------------------------------------------------------
__global__ __launch_bounds__(256) void k_gates(
    const float* __restrict__ x,
    const float* __restrict__ wi, const float* __restrict__ bi,
    const float* __restrict__ wf, const float* __restrict__ bf,
    const float* __restrict__ n_prev,
    const float* __restrict__ kvec, const float* __restrict__ qvec,
    float* __restrict__ n_out, float* __restrict__ scal)
{
    __shared__ float s1[256], s2[256];
    __shared__ float sh_i, sh_f;
    const int t = threadIdx.x;

    float pi = 0.f, pf = 0.f;
    for (int j = t; j < D; j += 256) {
        const float xv = x[j];
        pi = fmaf(wi[j], xv, pi);
        pf = fmaf(wf[j], xv, pf);
    }
    s1[t] = pi; s2[t] = pf;
    __syncthreads();
    for (int s = 128; s > 0; s >>= 1) {
        if (t < s) { s1[t] += s1[t + s]; s2[t] += s2[t + s]; }
        __syncthreads();
    }
    if (t == 0) {
        sh_i = __expf(s1[0] + bi[0]);
        sh_f = 1.0f / (1.0f + __expf(-(s2[0] + bf[0])));
        scal[0] = sh_i;
        scal[1] = sh_f;
    }
    __syncthreads();
    const float it = sh_i, ft = sh_f;

    float pn = 0.f;
    for (int j = t; j < D; j += 256) {
        const float nt = fmaf(ft, n_prev[j], it * kvec[j]);
        n_out[j] = nt;
        pn = fmaf(nt, qvec[j], pn);
    }
    s1[t] = pn;
    __syncthreads();
    for (int s = 128; s > 0; s >>= 1) {
        if (t < s) s1[t] += s1[t + s];
        __syncthreads();
    }
    if (t == 0) scal[2] = fmaxf(fabsf(s1[0]), 1.0f);
}

// ---------------------------------------------------------------------------
// Kernel 3: matrix-memory update fused with h_tilde GEMV.
// One wave per row r:  c_t[r,:] = f*c_prev[r,:] + (i*v[r])*k[:]  (written once)
//                      h~[r]    = c_t[r,:] . q  (accumulated in flight)
//                      h[r]     = o[r] * h~[r] / max_nqt
// float4 traffic everywhere; c_t is never re-read from memory.
// ---------------------------------------------------------------------------
__global__ __launch_bounds__(256) void k_cell(
    const float* __restrict__ c_prev,
    const float* __restrict__ kvec, const float* __restrict__ qvec,
    const float* __restrict__ vvec, const float* __restrict__ ovec,
    const float* __restrict__ scal,
    float* __restrict__ c_out, float* __restrict__ h_out)
{
    const int lane = threadIdx.x & 31;
    const int wave = threadIdx.x >> 5;
    const int r = blockIdx.x * 8 + wave;

    const float it = scal[0], ft = scal[1], mn = scal[2];
    const float coef = it * vvec[r];

    const float4* cp = (const float4*)(c_prev + (size_t)r * D);
    float4*       co = (float4*)(c_out + (size_t)r * D);
    const float4* k4 = (const float4*)kvec;
    const float4* q4 = (const float4*)qvec;

    float acc = 0.f;
    for (int j = lane; j < D / 4; j += 32) {
        const float4 cpv = cp[j];
        const float4 kk  = k4[j];
        const float4 qq  = q4[j];
        float4 ct;
        ct.x = fmaf(ft, cpv.x, coef * kk.x);
        ct.y = fmaf(ft, cpv.y, coef * kk.y);
        ct.z = fmaf(ft, cpv.z, coef * kk.z);
        ct.w = fmaf(ft, cpv.w, coef * kk.w);
        co[j] = ct;
        acc = fmaf(ct.x, qq.x, acc);
        acc = fmaf(ct.y, qq.y, acc);
        acc = fmaf(ct.z, qq.z, acc);
        acc = fmaf(ct.w, qq.w, acc);
    }
    // wave32 reduction
    #pragma unroll
    for (int off = 16; off > 0; off >>= 1)
        acc += __shfl_down(acc, off, 32);
    if (lane == 0)
        h_out[r] = ovec[r] * (acc / mn);
}

// ---------------------------------------------------------------------------
extern "C" void kernel_launch(void* const* d_in, const int* in_sizes, int n_in,
                              void* d_out, int out_size, void* d_ws, size_t ws_size,
                              hipStream_t stream)
{
    const float* x      = (const float*)d_in[0];
    const float* c_prev = (const float*)d_in[1];
    const float* n_prev = (const float*)d_in[2];
    const float* wq = (const float*)d_in[3];  const float* bq = (const float*)d_in[4];
    const float* wk = (const float*)d_in[5];  const float* bk = (const float*)d_in[6];
    const float* wv = (const float*)d_in[7];  const float* bv = (const float*)d_in[8];
    const float* wi = (const float*)d_in[9];  const float* bi = (const float*)d_in[10];
    const float* wf = (const float*)d_in[11]; const float* bf = (const float*)d_in[12];
    const float* wo = (const float*)d_in[13]; const float* bo = (const float*)d_in[14];

    // Output layout: h [D] | c_t [D*D] | n_t [D]
    float* h = (float*)d_out;
    float* c = h + D;
    float* n = c + (size_t)D * D;

    // Workspace: q | k | v | o | {i_t, f_t, max_nqt}
    float* q    = (float*)d_ws;
    float* kv   = q + D;
    float* v    = kv + D;
    float* o    = v + D;
    float* scal = o + D;

    dim3 g1(D / (16 * 4), 4);   // 64 row-tile groups x 4 matrices
    k_gemv4_wmma<<<g1, 128, 0, stream>>>(x, wq, bq, wk, bk, wv, bv, wo, bo,
                                         q, kv, v, o);
    k_gates<<<1, 256, 0, stream>>>(x, wi, bi, wf, bf, n_prev, kv, q, n, scal);
    k_cell<<<D / 8, 256, 0, stream>>>(c_prev, kv, q, v, o, scal, c, h);
}